// S4D_54528904790569
// MI455X (gfx1250) — compile-verified
//
#include <hip/hip_runtime.h>
#include <hip/hip_bf16.h>
#include <math.h>

// Problem constants (from reference): B=4, H=768, N2=32, L=4096
#define SB   4
#define SH   768
#define SN2  32
#define SL   4096

typedef __attribute__((ext_vector_type(2))) float    v2f;
typedef __attribute__((ext_vector_type(8))) float    v8f;
typedef __attribute__((ext_vector_type(4))) unsigned v4u;
typedef __attribute__((ext_vector_type(8))) unsigned v8u;

// ---------------------------------------------------------------------------
// Kernel 1: S4D recurrent scan (replaces Vandermonde + FFT conv), fused with
// D-skip and exact GELU. One wave32 per (b,h); lane n owns diagonal state n
// (N2 == 32 == warpSize). Per-timestep sum over n via xor-shuffle tree.
// ---------------------------------------------------------------------------
__global__ __launch_bounds__(256) void s4d_scan_kernel(
    const float* __restrict__ u,           // [B,H,L]
    const float* __restrict__ log_dt,      // [H]
    const float* __restrict__ Cri,         // [H,N2,2]
    const float* __restrict__ log_A_real,  // [H,N2]
    const float* __restrict__ A_imag,      // [H,N2]
    const float* __restrict__ Dv,          // [H]
    float* __restrict__ y_act)             // [B,H,L]
{
    const int lane = threadIdx.x & 31;
    const int wid  = (blockIdx.x * blockDim.x + threadIdx.x) >> 5;  // 0..B*H-1
    const int b    = wid / SH;
    const int h    = wid - b * SH;
    const int n    = lane;                       // diagonal state index

    // --- per-(h,n) constants: z = exp(dt*A), Cmod = C * (z-1)/A, fold 2x ---
    const float dt  = __expf(log_dt[h]);
    const float Ar  = -__expf(log_A_real[h * SN2 + n]);
    const float Ai  = A_imag[h * SN2 + n];
    float s, c;
    __sincosf(dt * Ai, &s, &c);
    const float mag = __expf(dt * Ar);
    const float zr  = mag * c;
    const float zi  = mag * s;
    const float inv = 1.0f / (Ar * Ar + Ai * Ai);
    const float qr  = ((zr - 1.0f) * Ar + zi * Ai) * inv;
    const float qi  = (zi * Ar - (zr - 1.0f) * Ai) * inv;
    const float Cr  = Cri[(h * SN2 + n) * 2 + 0];
    const float Ci  = Cri[(h * SN2 + n) * 2 + 1];
    const float Cmr = 2.0f * (Cr * qr - Ci * qi);
    const float Cmi = 2.0f * (Cr * qi + Ci * qr);
    const float Dh  = Dv[h];

    const size_t base = ((size_t)b * SH + h) * SL;
    float sr = 0.0f, si = 0.0f;                  // complex state

    for (int l0 = 0; l0 < SL; l0 += 32) {
        const float ubuf = u[base + l0 + lane];  // 32 timesteps staged in lanes
        if (l0 + 1024 < SL)                      // global_prefetch_b8
            __builtin_prefetch(&u[base + l0 + 1024], 0, 1);
        float yreg = 0.0f;
#pragma unroll
        for (int j = 0; j < 32; ++j) {
            const float uv = __shfl(ubuf, j, 32);            // v_readlane
            const float nsr = fmaf(zr, sr, fmaf(-zi, si, uv));
            const float nsi = fmaf(zr, si, zi * sr);
            sr = nsr; si = nsi;
            float cv = fmaf(Cmr, sr, -Cmi * si);
#pragma unroll
            for (int m = 16; m > 0; m >>= 1)
                cv += __shfl_xor(cv, m, 32);
            if (j == lane) yreg = cv;            // lane keeps "its" timestep
        }
        const float x = fmaf(Dh, ubuf, yreg);
        const float g = x * 0.5f * (1.0f + erff(x * 0.70710678118654752f));
        y_act[base + l0 + lane] = g;
    }
}

// ---------------------------------------------------------------------------
// TDM panel load: one tensor_load_to_lds DMAs a 768x16 f32 tile (16 floats
// per row, row stride L=4096) from global straight into LDS.  D# groups per
// ISA 8.3/8.4; block-uniform -> SGPR tuples; tracked with TENSORcnt.
// ---------------------------------------------------------------------------
__device__ __forceinline__ void tdm_load_panel(const float* gsrc, unsigned lds_off) {
    const unsigned long long ga = (unsigned long long)(uintptr_t)gsrc;
    v4u g0;
    g0[0] = 1u;                                            // count=1 valid D#
    g0[1] = lds_off;                                       // lds_addr (bytes)
    g0[2] = (unsigned)ga;                                  // global_addr[31:0]
    g0[3] = (unsigned)((ga >> 32) & 0x01FFFFFFu) | (2u << 30); // [56:32] | type=2
    v8u g1;
    g1[0] = (2u << 16);                                    // data_size = 4B
    g1[1] = (unsigned)((SL & 0xFFFFu) << 16);              // tensor_dim0 lo16
    g1[2] = (unsigned)((SL >> 16) & 0xFFFFu) | ((SH & 0xFFFFu) << 16); // td0 hi | td1 lo
    g1[3] = (unsigned)((SH >> 16) & 0xFFFFu) | (16u << 16);// td1 hi | tile_dim0=16
    g1[4] = (unsigned)SH;                                  // tile_dim1=768, tile_dim2=0
    g1[5] = (unsigned)SL;                                  // tensor_dim0_stride[31:0]
    g1[6] = 0u;                                            // stride0 hi | stride1 lo
    g1[7] = 0u;
    asm volatile("tensor_load_to_lds %0, %1" :: "s"(g0), "s"(g1) : "memory");
}

// ---------------------------------------------------------------------------
// Kernel 2: z = W_out[2H,H] @ y_act (+bias) -> GLU via V_WMMA_F32_16X16X4_F32.
// Persistent block over 2 column tiles with double-buffered TDM panel loads:
// tile t+1's DMA overlaps tile t's 384 WMMAs (TENSORcnt is in-order, so
// s_wait_tensorcnt 1 == "buffer0 ready").  8 waves = 8 h-tiles per block,
// each wave runs TWO f32 accumulators (GLU a/g halves) sharing the B operand.
// ---------------------------------------------------------------------------
__global__ __launch_bounds__(256) void s4d_gemm_glu_kernel(
    const float* __restrict__ y_act,   // [B,H,L]
    const float* __restrict__ W,       // [2H,H] row-major
    const float* __restrict__ bo,      // [2H]
    float* __restrict__ out)           // [B,H,L]
{
    __shared__ float ldsY[2][SH * 16]; // 2 x 48 KB double buffer

    const int lane = threadIdx.x & 31;
    const int wv   = threadIdx.x >> 5;             // wave 0..7 -> h tile
    const int h0   = (blockIdx.y * 8 + wv) * 16;
    const int nn   = lane & 15;                    // WMMA N (and A-matrix M)
    const int hi   = lane >> 4;                    // selects K pair {0,1}/{2,3}
    const float* Wa = W + (size_t)(h0 + nn) * SH;        // GLU 'a' row
    const float* Wg = W + (size_t)(h0 + nn + SH) * SH;   // GLU 'g' row

    const int ct0 = blockIdx.x * 2;                // two column tiles per block

    if (threadIdx.x < 32) {                        // prologue: DMA tile 0
        const int bi = ct0 >> 8, l0 = (ct0 & 255) << 4;
        tdm_load_panel(y_act + ((size_t)bi * SH) * SL + l0,
                       (unsigned)(uintptr_t)&ldsY[0][0]);
    }

    for (int it = 0; it < 2; ++it) {
        const int ct = ct0 + it;
        const int bi = ct >> 8;                    // L/16 = 256 tiles per batch
        const int l0 = (ct & 255) << 4;

        if (threadIdx.x < 32) {
            if (it == 0) {                         // issue next DMA, wait current
                const int cn = ct0 + 1;
                tdm_load_panel(y_act + ((size_t)(cn >> 8) * SH) * SL + ((cn & 255) << 4),
                               (unsigned)(uintptr_t)&ldsY[1][0]);
                __builtin_amdgcn_s_wait_tensorcnt(1);
            } else {
                __builtin_amdgcn_s_wait_tensorcnt(0);
            }
        }
        __syncthreads();
        const float* panel = &ldsY[it][0];

        v8f acc_a = {0.f, 0.f, 0.f, 0.f, 0.f, 0.f, 0.f, 0.f};
        v8f acc_g = {0.f, 0.f, 0.f, 0.f, 0.f, 0.f, 0.f, 0.f};

        for (int kk = 0; kk < SH; kk += 4) {
            // A: 16x4 f32, lane holds (M=nn, K = 2*hi + {0,1}) -> b64 load
            const v2f av = *(const v2f*)(Wa + kk + 2 * hi);
            const v2f gv = *(const v2f*)(Wg + kk + 2 * hi);
            // B: 4x16 f32 from LDS, vgpr0: K=hi, vgpr1: K=2+hi, col nn
            v2f bv;
            bv.x = panel[(kk + hi) * 16 + nn];
            bv.y = panel[(kk + 2 + hi) * 16 + nn];
            acc_a = __builtin_amdgcn_wmma_f32_16x16x4_f32(
                false, av, false, bv, (short)0, acc_a, false, false);
            acc_g = __builtin_amdgcn_wmma_f32_16x16x4_f32(
                false, gv, false, bv, (short)0, acc_g, false, false);
        }

        // epilogue: bias + GLU; D-matrix vgpr j -> row h0 + j + 8*hi
        const size_t obase = ((size_t)bi * SH) * SL + l0 + nn;
#pragma unroll
        for (int j = 0; j < 8; ++j) {
            const int row = h0 + j + 8 * hi;
            const float avv = acc_a[j] + bo[row];
            const float gvv = acc_g[j] + bo[row + SH];
            const float sg  = 1.0f / (1.0f + __expf(-gvv));
            out[obase + (size_t)row * SL] = avv * sg;
        }
    }
}

// ---------------------------------------------------------------------------
extern "C" void kernel_launch(void* const* d_in, const int* in_sizes, int n_in,
                              void* d_out, int out_size, void* d_ws, size_t ws_size,
                              hipStream_t stream) {
    const float* u          = (const float*)d_in[0];
    const float* log_dt     = (const float*)d_in[1];
    const float* C          = (const float*)d_in[2];
    const float* log_A_real = (const float*)d_in[3];
    const float* A_imag     = (const float*)d_in[4];
    const float* D          = (const float*)d_in[5];
    const float* W_out      = (const float*)d_in[6];
    const float* b_out      = (const float*)d_in[7];
    float* out  = (float*)d_out;
    float* yact = (float*)d_ws;                    // [B,H,L] f32 = 48 MB

    // 1) recurrent S4D scan + skip + GELU: B*H waves, 8 waves/block
    s4d_scan_kernel<<<dim3((SB * SH) / 8), dim3(256), 0, stream>>>(
        u, log_dt, C, log_A_real, A_imag, D, yact);

    // 2) WMMA f32 GEMM + bias + GLU: 512 blocks x 2 tiles (double-buffered TDM)
    s4d_gemm_glu_kernel<<<dim3((SB * SL) / 32, SH / (16 * 8)), dim3(256), 0, stream>>>(
        yact, W_out, b_out, out);
}